// AttentionAdjacency_49280454754755
// MI455X (gfx1250) — compile-verified
//
#include <hip/hip_runtime.h>
#include <cmath>

typedef __attribute__((ext_vector_type(2))) float v2f;
typedef __attribute__((ext_vector_type(8))) float v8f;

constexpr int B_ = 8, N_ = 1024, D_ = 256, H_ = 4, HD_ = 64, TOPK_ = 32;
constexpr int LDSS = 1032;  // padded LDS row stride (floats) to avoid bank conflicts

// D = A(16x4) * B(4x16) + C, full fp32. 8-arg form mirrors the confirmed f16 variant.
#define WMMA_F32_4(a, b, c) \
    __builtin_amdgcn_wmma_f32_16x16x4_f32(false, (a), false, (b), (short)0, (c), false, false)

__device__ __forceinline__ float waveMax(float x) {
    #pragma unroll
    for (int o = 16; o > 0; o >>= 1) x = fmaxf(x, __shfl_xor(x, o, 32));
    return x;
}
__device__ __forceinline__ float waveSum(float x) {
    #pragma unroll
    for (int o = 16; o > 0; o >>= 1) x += __shfl_xor(x, o, 32);
    return x;
}

// ---------------- Kernel 1: Q/K/V projections (x @ W^T + b), head-split layout -------------
__global__ __launch_bounds__(32) void qkv_proj_kernel(
    const float* __restrict__ h,
    const float* __restrict__ Wq, const float* __restrict__ bq,
    const float* __restrict__ Wk, const float* __restrict__ bk,
    const float* __restrict__ Wv, const float* __restrict__ bv,
    float* __restrict__ Q, float* __restrict__ K, float* __restrict__ V) {
    const int rowblk  = blockIdx.x;   // over B*N/16 = 512
    const int coltile = blockIdx.y;   // over D/16   = 16
    const int which   = blockIdx.z;   // 0=Q 1=K 2=V
    const float* W    = which == 0 ? Wq : (which == 1 ? Wk : Wv);
    const float* bias = which == 0 ? bq : (which == 1 ? bk : bv);
    float* dst        = which == 0 ? Q  : (which == 1 ? K  : V);

    const int lane = threadIdx.x & 31;
    const int half = lane >> 4;       // K sub-split per ISA A/B f32 layout
    const int mn   = lane & 15;       // M for A-frag, N for B-frag
    const int row0 = rowblk * 16;
    const int col0 = coltile * 16;

    const float* aptr = h + (size_t)(row0 + mn) * D_;
    const float* bptr = W + (size_t)(col0 + mn) * D_;   // out[m][c] = sum_k h[m][k]*W[c][k]

    v8f acc = {};
    for (int kk = 0; kk < D_; kk += 4) {
        v2f a, b;
        a.x = aptr[kk + half * 2]; a.y = aptr[kk + half * 2 + 1];
        b.x = bptr[kk + half * 2]; b.y = bptr[kk + half * 2 + 1];
        acc = WMMA_F32_4(a, b, acc);
    }

    const int col  = col0 + mn;
    const int head = col >> 6, hd = col & 63;
    const float bb = bias[col];
    #pragma unroll
    for (int r = 0; r < 8; ++r) {
        const int m   = r + half * 8;
        const int row = row0 + m;
        const int bi  = row >> 10, nn = row & (N_ - 1);
        dst[(((size_t)(bi * H_ + head)) * N_ + nn) * HD_ + hd] = acc[r] + bb;
    }
}

// ---------------- Kernel 2: fused scores -> topk mask -> softmax -> adjacency + attn@V -----
__global__ __launch_bounds__(512) void attn_fused_kernel(
    const float* __restrict__ Q, const float* __restrict__ K, const float* __restrict__ V,
    const float* __restrict__ td, float* __restrict__ adj, float* __restrict__ msg) {
    extern __shared__ float sc[];                // 16 rows x LDSS floats
    const int b    = blockIdx.y;
    const int row0 = blockIdx.x * 16;
    const int tid  = threadIdx.x;
    const int wave = tid >> 5, lane = tid & 31;
    const int half = lane >> 4, mn = lane & 15;

    float adjAcc[32];
    #pragma unroll
    for (int j = 0; j < 32; ++j) adjAcc[j] = 0.f;

    for (int hh = 0; hh < H_; ++hh) {
        const float* Qh = Q + (size_t)(b * H_ + hh) * N_ * HD_;
        const float* Kh = K + (size_t)(b * H_ + hh) * N_ * HD_;
        const float* Vh = V + (size_t)(b * H_ + hh) * N_ * HD_;

        // ---- Stage A: S = (Q K^T)/8 - 0.1*td  -> LDS.  Each wave: 4 col-tiles of 16.
        {
            const float* aptr = Qh + (size_t)(row0 + mn) * HD_;
            const float* bptr[4];
            #pragma unroll
            for (int t = 0; t < 4; ++t)
                bptr[t] = Kh + (size_t)((wave * 4 + t) * 16 + mn) * HD_;
            v8f acc[4] = {{}, {}, {}, {}};
            for (int kk = 0; kk < HD_; kk += 4) {
                v2f a;
                a.x = aptr[kk + half * 2]; a.y = aptr[kk + half * 2 + 1];
                #pragma unroll
                for (int t = 0; t < 4; ++t) {
                    v2f bb;
                    bb.x = bptr[t][kk + half * 2]; bb.y = bptr[t][kk + half * 2 + 1];
                    acc[t] = WMMA_F32_4(a, bb, acc[t]);
                }
            }
            #pragma unroll
            for (int t = 0; t < 4; ++t) {
                const int col = (wave * 4 + t) * 16 + mn;
                #pragma unroll
                for (int r = 0; r < 8; ++r) {
                    const int m = r + half * 8;
                    const float s = acc[t][r] * 0.125f
                                  - 0.1f * td[((size_t)b * N_ + row0 + m) * N_ + col];
                    sc[m * LDSS + col] = s;
                }
            }
        }
        __syncthreads();

        // ---- Stage B: exact top-32 threshold + masked softmax.  Wave r owns row r.
        {
            float v[32], w[32];
            #pragma unroll
            for (int j = 0; j < 32; ++j) { v[j] = sc[wave * LDSS + j * 32 + lane]; w[j] = v[j]; }

            float rowmax = -INFINITY, kth = -INFINITY;
            for (int it = 0; it < TOPK_; ++it) {
                float lm = -INFINITY;
                #pragma unroll
                for (int j = 0; j < 32; ++j) lm = fmaxf(lm, w[j]);
                const float m = waveMax(lm);
                if (it == 0) rowmax = m;
                kth = m;                                   // after loop: 32nd largest (w/ dups)
                const unsigned long long bal = __ballot(lm == m);
                const int src = __ffsll(bal) - 1;
                if (lane == src) {                          // remove exactly one instance
                    bool rm = false;
                    #pragma unroll
                    for (int j = 0; j < 32; ++j)
                        if (!rm && w[j] == m) { w[j] = -INFINITY; rm = true; }
                }
            }

            float ssum = 0.f;
            #pragma unroll
            for (int j = 0; j < 32; ++j) {
                const float e = (v[j] >= kth) ? __expf(v[j] - rowmax) : 0.f;
                w[j] = e; ssum += e;
            }
            ssum = waveSum(ssum);                          // >= 1, row max always kept
            const float inv = 1.0f / ssum;
            #pragma unroll
            for (int j = 0; j < 32; ++j) {
                const float a = w[j] * inv;
                sc[wave * LDSS + j * 32 + lane] = a;       // attn back to LDS for Stage C
                adjAcc[j] += 0.25f * a;                    // mean over H=4 heads
            }
        }
        __syncthreads();

        // ---- Stage C: M(16x64) = attn(16x1024) @ V(1024x64).  Waves 0..3, one 16-col tile each.
        if (wave < 4) {
            const int n0 = wave * 16;
            v8f acc = {};
            for (int kk = 0; kk < N_; kk += 4) {
                const int k0 = kk + half * 2;
                v2f a, bb;
                a.x  = sc[mn * LDSS + k0];         a.y  = sc[mn * LDSS + k0 + 1];
                bb.x = Vh[(size_t)k0 * HD_ + n0 + mn];
                bb.y = Vh[(size_t)(k0 + 1) * HD_ + n0 + mn];
                acc = WMMA_F32_4(a, bb, acc);
            }
            #pragma unroll
            for (int r = 0; r < 8; ++r) {
                const int m = r + half * 8;
                msg[((size_t)b * N_ + row0 + m) * D_ + hh * HD_ + n0 + mn] = acc[r];
            }
        }
        __syncthreads();
    }

    // ---- adjacency out: wave r -> row r, coalesced stores
    float* dst = adj + ((size_t)b * N_ + row0 + wave) * N_;
    #pragma unroll
    for (int j = 0; j < 32; ++j) dst[j * 32 + lane] = adjAcc[j];
}

// ---------------- Kernel 3: messages = msg @ Wo^T + bo ------------------------------------
__global__ __launch_bounds__(32) void out_proj_kernel(
    const float* __restrict__ X, const float* __restrict__ Wo, const float* __restrict__ bo,
    float* __restrict__ out) {
    const int row0 = blockIdx.x * 16;
    const int col0 = blockIdx.y * 16;
    const int lane = threadIdx.x & 31;
    const int half = lane >> 4;
    const int mn   = lane & 15;

    const float* aptr = X  + (size_t)(row0 + mn) * D_;
    const float* bptr = Wo + (size_t)(col0 + mn) * D_;

    v8f acc = {};
    for (int kk = 0; kk < D_; kk += 4) {
        v2f a, b;
        a.x = aptr[kk + half * 2]; a.y = aptr[kk + half * 2 + 1];
        b.x = bptr[kk + half * 2]; b.y = bptr[kk + half * 2 + 1];
        acc = WMMA_F32_4(a, b, acc);
    }
    const float bb = bo[col0 + mn];
    #pragma unroll
    for (int r = 0; r < 8; ++r) {
        const int m = r + half * 8;
        out[(size_t)(row0 + m) * D_ + col0 + mn] = acc[r] + bb;
    }
}

extern "C" void kernel_launch(void* const* d_in, const int* in_sizes, int n_in,
                              void* d_out, int out_size, void* d_ws, size_t ws_size,
                              hipStream_t stream) {
    const float* h  = (const float*)d_in[0];
    const float* td = (const float*)d_in[1];
    const float* Wq = (const float*)d_in[2];
    const float* bq = (const float*)d_in[3];
    const float* Wk = (const float*)d_in[4];
    const float* bk = (const float*)d_in[5];
    const float* Wv = (const float*)d_in[6];
    const float* bv = (const float*)d_in[7];
    const float* Wo = (const float*)d_in[8];
    const float* bo = (const float*)d_in[9];
    // d_in[10] = top_k (fixed 32, matches TOPK_)

    float* ws = (float*)d_ws;
    const size_t QKV = (size_t)B_ * H_ * N_ * HD_;     // 2M floats each
    float* Q   = ws;
    float* K   = ws + QKV;
    float* V   = ws + 2 * QKV;
    float* MSG = ws + 3 * QKV;                          // B*N*D

    float* adj      = (float*)d_out;                    // (B,N,N) first
    float* msgs_out = (float*)d_out + (size_t)B_ * N_ * N_;

    dim3 g1(B_ * N_ / 16, D_ / 16, 3);
    qkv_proj_kernel<<<g1, 32, 0, stream>>>(h, Wq, bq, Wk, bk, Wv, bv, Q, K, V);

    dim3 g2(N_ / 16, B_);
    attn_fused_kernel<<<g2, 512, 16 * LDSS * sizeof(float), stream>>>(Q, K, V, td, adj, MSG);

    dim3 g3(B_ * N_ / 16, D_ / 16);
    out_proj_kernel<<<g3, 32, 0, stream>>>(MSG, Wo, bo, msgs_out);
}